// GraphLayer_69569880260772
// MI455X (gfx1250) — compile-verified
//
#include <hip/hip_runtime.h>
#include <hip/hip_bf16.h>

typedef __attribute__((ext_vector_type(2))) float v2f;
typedef __attribute__((ext_vector_type(8))) float v8f;

// ---------------------------------------------------------------------------
// CSR build: histogram -> single-block scan -> slot fill
// ---------------------------------------------------------------------------
__global__ void zero_cnt_kernel(int* cnt, int n) {
    int i = blockIdx.x * blockDim.x + threadIdx.x;
    if (i < n) cnt[i] = 0;
}

__global__ void count_kernel(const int* __restrict__ dst, int* cnt, int e) {
    int i = blockIdx.x * blockDim.x + threadIdx.x;
    if (i < e) atomicAdd(&cnt[dst[i]], 1);
}

// Single block, 1024 threads: exclusive scan of cnt -> rowstart & cursor,
// plus dinv[i] = rsqrt(cnt[i] + 1)  (self-loop included in degree).
__global__ void scan_kernel(const int* __restrict__ cnt, int* rowstart, int* cursor,
                            float* dinv, int n) {
    __shared__ int sh[1024];
    __shared__ int carry_sh;
    if (threadIdx.x == 0) carry_sh = 0;
    __syncthreads();
    const int chunks = (n + 1023) / 1024;
    for (int c = 0; c < chunks; ++c) {
        int i = c * 1024 + threadIdx.x;
        int v = (i < n) ? cnt[i] : 0;
        sh[threadIdx.x] = v;
        __syncthreads();
        for (int off = 1; off < 1024; off <<= 1) {
            int t = (threadIdx.x >= off) ? sh[threadIdx.x - off] : 0;
            __syncthreads();
            sh[threadIdx.x] += t;
            __syncthreads();
        }
        int incl = sh[threadIdx.x];
        int excl = incl - v + carry_sh;          // carry from previous chunk
        if (i < n) {
            rowstart[i] = excl;
            cursor[i]   = excl;
            dinv[i]     = rsqrtf((float)(v + 1));
        }
        __syncthreads();
        if (threadIdx.x == 1023) carry_sh += incl;   // chunk total
        __syncthreads();
    }
    if (threadIdx.x == 0) rowstart[n] = carry_sh;    // == E
}

__global__ void fill_kernel(const int* __restrict__ src, const int* __restrict__ dst,
                            int* cursor, int* colidx, int e) {
    int i = blockIdx.x * blockDim.x + threadIdx.x;
    if (i < e) {
        int d = dst[i];
        int p = atomicAdd(&cursor[d], 1);
        colidx[p] = src[i];
    }
}

// ---------------------------------------------------------------------------
// fp32 WMMA GEMM:  Out[M x Ncols] = A[M x K] @ W[Ncols x K]^T (+ bias)
// 8 waves / 256-thread block; each wave computes TWO adjacent 16x16 tiles
// sharing one A fragment per K-step. Block covers 16 rows x 256 cols.
// Grid = (M/16, ceil(Ncols/256)).
// V_WMMA_F32_16X16X4_F32 layouts (ISA 7.12.2):
//   A 16x4: lanes 0-15 -> M=lane, K={0,1}; lanes 16-31 -> M=lane-16, K={2,3}
//   B 4x16: lanes 0-15 -> N=lane, K={0,1}; lanes 16-31 -> K={2,3}
//   D 16x16: VGPR r, lanes 0-15 -> row r, col=lane; lanes 16-31 -> row r+8
// ---------------------------------------------------------------------------
__global__ void gemm_wmma_f32_kernel(const float* __restrict__ A,
                                     const float* __restrict__ W,
                                     const float* __restrict__ bias,
                                     float* __restrict__ Out,
                                     int Ncols, int K) {
    const int lane = threadIdx.x & 31;
    const int wave = threadIdx.x >> 5;                 // 0..7
    const int row0 = blockIdx.x * 16;
    const int col0 = (blockIdx.y * 8 + wave) * 32;     // two 16-wide tiles
    if (col0 >= Ncols) return;                         // wave-uniform: EXEC stays all-ones

    const int fm = lane & 15;
    const int kb = (lane >> 4) << 1;                   // 0 | 2

    const float* arow = A + (size_t)(row0 + fm) * K + kb;
    const float* w0   = W + (size_t)(col0 + fm) * K + kb;        // B0[k][n] = W[col0+n][k]
    const float* w1   = W + (size_t)(col0 + 16 + fm) * K + kb;   // B1

    v8f acc0 = {}, acc1 = {};
    #pragma unroll 4
    for (int k = 0; k < K; k += 4) {
        v2f a  = *(const v2f*)(arow + k);
        v2f b0 = *(const v2f*)(w0 + k);
        v2f b1 = *(const v2f*)(w1 + k);
        acc0 = __builtin_amdgcn_wmma_f32_16x16x4_f32(false, a, false, b0,
                                                     (short)0, acc0, false, false);
        acc1 = __builtin_amdgcn_wmma_f32_16x16x4_f32(false, a, false, b1,
                                                     (short)0, acc1, false, false);
    }

    const int n    = lane & 15;
    const int mrow = (lane >> 4) * 8;
    const float bv0 = bias ? bias[col0 + n]      : 0.0f;
    const float bv1 = bias ? bias[col0 + 16 + n] : 0.0f;
    #pragma unroll
    for (int r = 0; r < 8; ++r) {
        float* orow = Out + (size_t)(row0 + mrow + r) * Ncols;
        orow[col0 + n]      = acc0[r] + bv0;
        orow[col0 + 16 + n] = acc1[r] + bv1;
    }
}

// ---------------------------------------------------------------------------
// Pull-based aggregation (atomic-free hot path):
//   out[d,f] = dinv[d] * ( dinv[d]*h[d,f] + sum_{s in N(d)} dinv[s]*h[s,f] ) + bias[f]
// One block per node, one thread per feature (blockDim == D). Coalesced row reads.
// ---------------------------------------------------------------------------
__global__ void gather_kernel(const float* __restrict__ h,
                              const int* __restrict__ rowstart,
                              const int* __restrict__ colidx,
                              const float* __restrict__ dinv,
                              const float* __restrict__ bias,
                              float* __restrict__ out, int d) {
    const int node = blockIdx.x;
    const int f = threadIdx.x;
    const float dn = dinv[node];
    float acc = dn * h[(size_t)node * d + f];
    const int beg = rowstart[node];
    const int end = rowstart[node + 1];
    for (int j = beg; j < end; ++j) {
        int s = colidx[j];                     // uniform per block -> broadcast
        acc += dinv[s] * h[(size_t)s * d + f]; // coalesced 1KB row read
    }
    out[(size_t)node * d + f] = dn * acc + bias[f];
}

// ---------------------------------------------------------------------------
// Tiny head: out[m, 0..C-1] = tmp[m,:] @ Wm2^T + bm2   (C=5, K=128)
// ---------------------------------------------------------------------------
__global__ void mlp_out_kernel(const float* __restrict__ tmp,
                               const float* __restrict__ Wm2,
                               const float* __restrict__ bm2,
                               float* __restrict__ out,
                               int n, int hh, int c) {
    int m = blockIdx.x * blockDim.x + threadIdx.x;
    if (m >= n) return;
    const float* t = tmp + (size_t)m * hh;
    for (int j = 0; j < c; ++j) {
        const float* w = Wm2 + (size_t)j * hh;
        float s = bm2[j];
        for (int k = 0; k < hh; ++k) s += t[k] * w[k];
        out[(size_t)m * c + j] = s;
    }
}

// ---------------------------------------------------------------------------
// Launch
// ---------------------------------------------------------------------------
static inline char* align256(char* p) {
    return (char*)(((uintptr_t)p + 255u) & ~(uintptr_t)255u);
}

extern "C" void kernel_launch(void* const* d_in, const int* in_sizes, int n_in,
                              void* d_out, int out_size, void* d_ws, size_t ws_size,
                              hipStream_t stream) {
    const float* x   = (const float*)d_in[0];
    const int*   ei  = (const int*)d_in[1];
    const float* W1  = (const float*)d_in[2];
    const float* b1  = (const float*)d_in[3];
    const float* W2  = (const float*)d_in[4];
    const float* b2  = (const float*)d_in[5];
    const float* Wm1 = (const float*)d_in[6];
    const float* bm1 = (const float*)d_in[7];
    const float* Wm2 = (const float*)d_in[8];
    const float* bm2 = (const float*)d_in[9];

    const int D  = in_sizes[3];            // 256
    const int Hh = in_sizes[7];            // 128
    const int C  = in_sizes[9];            // 5
    const int N  = in_sizes[0] / D;        // 100000
    const int E  = in_sizes[1] / 2;        // 3.2M

    const int* src = ei;                   // edge_index[0]
    const int* dst = ei + E;               // edge_index[1]

    // workspace layout (256B aligned blocks):
    // dinv[N] | cnt[N] | rowstart[N+1] | cursor[N] | colidx[E] | Hbuf[N*D] | Gbuf[N*D]
    char* p = (char*)d_ws;
    float* dinv    = (float*)p;            p = align256(p + sizeof(float) * N);
    int* cnt       = (int*)p;              p = align256(p + sizeof(int) * N);
    int* rowstart  = (int*)p;              p = align256(p + sizeof(int) * (size_t)(N + 1));
    int* cursor    = (int*)p;              p = align256(p + sizeof(int) * N);
    int* colidx    = (int*)p;              p = align256(p + sizeof(int) * (size_t)E);
    float* Hbuf    = (float*)p;            p = align256(p + sizeof(float) * (size_t)N * D);
    float* Gbuf    = (float*)p;

    float* out = (float*)d_out;

    const int TB = 256;
    dim3 blk(TB);

    // --- CSR build + dinv ---
    zero_cnt_kernel<<<dim3((N + TB - 1) / TB), blk, 0, stream>>>(cnt, N);
    count_kernel<<<dim3((E + TB - 1) / TB), blk, 0, stream>>>(dst, cnt, E);
    scan_kernel<<<dim3(1), dim3(1024), 0, stream>>>(cnt, rowstart, cursor, dinv, N);
    fill_kernel<<<dim3((E + TB - 1) / TB), blk, 0, stream>>>(src, dst, cursor, colidx, E);

    const int mtiles = N / 16;             // 6250, exact
    dim3 ggrid((unsigned)N);               // one block per node
    dim3 gblk((unsigned)D);                // one thread per feature

    // --- layer 1 ---
    gemm_wmma_f32_kernel<<<dim3(mtiles, (D + 255) / 256), blk, 0, stream>>>(x, W1, nullptr, Hbuf, D, D);
    gather_kernel<<<ggrid, gblk, 0, stream>>>(Hbuf, rowstart, colidx, dinv, b1, Gbuf, D);

    // --- layer 2 ---
    gemm_wmma_f32_kernel<<<dim3(mtiles, (D + 255) / 256), blk, 0, stream>>>(Gbuf, W2, nullptr, Hbuf, D, D);
    gather_kernel<<<ggrid, gblk, 0, stream>>>(Hbuf, rowstart, colidx, dinv, b2, Gbuf, D);

    // --- MLP hidden: tmp = agg2 @ Wm1^T + bm1 (reuse Hbuf, N x 128) ---
    gemm_wmma_f32_kernel<<<dim3(mtiles, (Hh + 255) / 256), blk, 0, stream>>>(Gbuf, Wm1, bm1, Hbuf, Hh, D);
    // --- head ---
    mlp_out_kernel<<<dim3((N + TB - 1) / TB), blk, 0, stream>>>(Hbuf, Wm2, bm2, out, N, Hh, C);
}